// PointDeformer_82841329205495
// MI455X (gfx1250) — compile-verified
//
#include <hip/hip_runtime.h>
#include <stdint.h>

typedef __attribute__((ext_vector_type(2))) float v2f;
typedef __attribute__((ext_vector_type(8))) float v8f;

#define NJ   24
#define NVTX 6890

__constant__ int c_parents[NJ] = {-1,0,0,0,1,2,3,4,5,6,7,8,9,9,9,12,13,14,16,17,18,19,20,21};

// ---------------------------------------------------------------------------
// Workspace layout (floats):
//   [0   ..  72)  J0  (24x3)  rest joints (beta = 0)
//   [72  .. 144)  J1  (24x3)  posed-shape joints (with beta)
//   [144 .. 528)  WMMA B-operand staging: 6 chunks x 2 VGPRs x 32 lanes
//   [528 .. 816)  A (24x12) flattened row-major 3x4 transforms (tail path)
// ---------------------------------------------------------------------------

// Kernel 1: J0[j] = Jreg[j,:] @ v_template ; J1[j] = Jreg[j,:] @ (v_template + shapedirs.beta)
__global__ void smpl_joints_kernel(const float* __restrict__ shapedirs,
                                   const float* __restrict__ v_template,
                                   const float* __restrict__ Jreg,
                                   const float* __restrict__ beta,
                                   float* __restrict__ ws)
{
    const int j   = blockIdx.x;   // joint
    const int tid = threadIdx.x;  // 0..255
    float b[10];
#pragma unroll
    for (int k = 0; k < 10; ++k) b[k] = beta[k];

    float acc[6] = {0.f,0.f,0.f,0.f,0.f,0.f};
    for (int v = tid; v < NVTX; v += blockDim.x) {
        float w = Jreg[j * NVTX + v];
#pragma unroll
        for (int c = 0; c < 3; ++c) {
            float vt = v_template[v * 3 + c];
            const float* sd = shapedirs + ((size_t)v * 3 + c) * 10;
            float s = vt;
#pragma unroll
            for (int k = 0; k < 10; ++k) s += sd[k] * b[k];
            acc[c]     += w * vt;  // beta = 0 path
            acc[3 + c] += w * s;   // beta path
        }
    }

    __shared__ float sm[256];
#pragma unroll
    for (int i = 0; i < 6; ++i) {
        sm[tid] = acc[i];
        __syncthreads();
        for (int s = 128; s > 0; s >>= 1) {
            if (tid < s) sm[tid] += sm[tid + s];
            __syncthreads();
        }
        if (tid == 0) {
            if (i < 3) ws[j * 3 + i]            = sm[0];
            else       ws[72 + j * 3 + (i - 3)] = sm[0];
        }
        __syncthreads();
    }
}

__device__ inline void rodrigues3(float rx, float ry, float rz, float R[9])
{
    float n  = sqrtf(rx * rx + ry * ry + rz * rz) + 1e-8f;  // matches reference eps
    float ux = rx / n, uy = ry / n, uz = rz / n;
    float c = cosf(n), s = sinf(n), o = 1.f - c;
    R[0] = c + o * ux * ux;     R[1] = o * ux * uy - s * uz; R[2] = o * ux * uz + s * uy;
    R[3] = o * uy * ux + s * uz; R[4] = c + o * uy * uy;     R[5] = o * uy * uz - s * ux;
    R[6] = o * uz * ux - s * uy; R[7] = o * uz * uy + s * ux; R[8] = c + o * uz * uz;
}

// C = A o B  (3x4 affine, implicit bottom row [0,0,0,1])
__device__ inline void affmul(const float* A, const float* B, float* C)
{
#pragma unroll
    for (int i = 0; i < 3; ++i) {
#pragma unroll
        for (int k = 0; k < 3; ++k)
            C[i*4+k] = A[i*4+0]*B[0*4+k] + A[i*4+1]*B[1*4+k] + A[i*4+2]*B[2*4+k];
        C[i*4+3] = A[i*4+0]*B[3] + A[i*4+1]*B[7] + A[i*4+2]*B[11] + A[i*4+3];
    }
}

// Kernel 2: build A[j] = pose_j @ inv(da_j) and the WMMA B-operand staging.
__global__ void smpl_transforms_kernel(const float* __restrict__ theta,
                                       const float* __restrict__ da_theta,
                                       float* __restrict__ ws)
{
    __shared__ float Lp[NJ][12];   // local pose transforms
    __shared__ float Ld[NJ][12];   // local rest ("da") transforms
    __shared__ float Afin[NJ][12];
    const int tid = threadIdx.x;
    const float* J0 = ws;        // rest joints
    const float* J1 = ws + 72;   // posed-shape joints

    if (tid < NJ) {
        const int j = tid;
        const int p = c_parents[j];
        float R[9];
        // pose local
        rodrigues3(theta[j*3], theta[j*3+1], theta[j*3+2], R);
        float tx = J1[j*3], ty = J1[j*3+1], tz = J1[j*3+2];
        if (p >= 0) { tx -= J1[p*3]; ty -= J1[p*3+1]; tz -= J1[p*3+2]; }
        Lp[j][0]=R[0]; Lp[j][1]=R[1]; Lp[j][2]=R[2]; Lp[j][3]=tx;
        Lp[j][4]=R[3]; Lp[j][5]=R[4]; Lp[j][6]=R[5]; Lp[j][7]=ty;
        Lp[j][8]=R[6]; Lp[j][9]=R[7]; Lp[j][10]=R[8]; Lp[j][11]=tz;
        // rest local
        rodrigues3(da_theta[j*3], da_theta[j*3+1], da_theta[j*3+2], R);
        tx = J0[j*3]; ty = J0[j*3+1]; tz = J0[j*3+2];
        if (p >= 0) { tx -= J0[p*3]; ty -= J0[p*3+1]; tz -= J0[p*3+2]; }
        Ld[j][0]=R[0]; Ld[j][1]=R[1]; Ld[j][2]=R[2]; Ld[j][3]=tx;
        Ld[j][4]=R[3]; Ld[j][5]=R[4]; Ld[j][6]=R[5]; Ld[j][7]=ty;
        Ld[j][8]=R[6]; Ld[j][9]=R[7]; Ld[j][10]=R[8]; Ld[j][11]=tz;
    }
    __syncthreads();

    if (tid < NJ) {
        const int j = tid;
        // ancestor path root..j
        int path[NJ]; int d = 0; int cur = j;
        while (cur >= 0) { path[d++] = cur; cur = c_parents[cur]; }
        float Gp[12], Gd[12], tmp[12];
#pragma unroll
        for (int i = 0; i < 12; ++i) { Gp[i] = Lp[path[d-1]][i]; Gd[i] = Ld[path[d-1]][i]; }
        for (int i = d - 2; i >= 0; --i) {
            affmul(Gp, &Lp[path[i]][0], tmp);
            for (int k = 0; k < 12; ++k) Gp[k] = tmp[k];
            affmul(Gd, &Ld[path[i]][0], tmp);
            for (int k = 0; k < 12; ++k) Gd[k] = tmp[k];
        }
        // "pack" step: last column -= R_G @ J
        {
            float jx = J1[j*3], jy = J1[j*3+1], jz = J1[j*3+2];
            Gp[3]  -= Gp[0]*jx + Gp[1]*jy + Gp[2]*jz;
            Gp[7]  -= Gp[4]*jx + Gp[5]*jy + Gp[6]*jz;
            Gp[11] -= Gp[8]*jx + Gp[9]*jy + Gp[10]*jz;
            jx = J0[j*3]; jy = J0[j*3+1]; jz = J0[j*3+2];
            Gd[3]  -= Gd[0]*jx + Gd[1]*jy + Gd[2]*jz;
            Gd[7]  -= Gd[4]*jx + Gd[5]*jy + Gd[6]*jz;
            Gd[11] -= Gd[8]*jx + Gd[9]*jy + Gd[10]*jz;
        }
        // invert rest transform (affine: invert 3x3 via adjugate, t' = -Minv t)
        float m00=Gd[0], m01=Gd[1], m02=Gd[2];
        float m10=Gd[4], m11=Gd[5], m12=Gd[6];
        float m20=Gd[8], m21=Gd[9], m22=Gd[10];
        float det = m00*(m11*m22-m12*m21) - m01*(m10*m22-m12*m20) + m02*(m10*m21-m11*m20);
        float id = 1.0f / det;
        float i00 =  (m11*m22-m12*m21)*id, i01 = -(m01*m22-m02*m21)*id, i02 =  (m01*m12-m02*m11)*id;
        float i10 = -(m10*m22-m12*m20)*id, i11 =  (m00*m22-m02*m20)*id, i12 = -(m00*m12-m02*m10)*id;
        float i20 =  (m10*m21-m11*m20)*id, i21 = -(m00*m21-m01*m20)*id, i22 =  (m00*m11-m01*m10)*id;
        float itx = -(i00*Gd[3] + i01*Gd[7] + i02*Gd[11]);
        float ity = -(i10*Gd[3] + i11*Gd[7] + i12*Gd[11]);
        float itz = -(i20*Gd[3] + i21*Gd[7] + i22*Gd[11]);
        float Di[12] = { i00,i01,i02,itx, i10,i11,i12,ity, i20,i21,i22,itz };
        float A[12];
        affmul(Gp, Di, A);
#pragma unroll
        for (int i = 0; i < 12; ++i) { Afin[j][i] = A[i]; ws[528 + j*12 + i] = A[i]; }
    }
    __syncthreads();

    // WMMA B-operand staging: 4x16 fp32 B layout -> VGPR0: K0 (lanes0-15)/K2 (lanes16-31),
    // VGPR1: K1/K3; lane = output column.
    float* staged = ws + 144;
    for (int e = tid; e < 6 * 64; e += blockDim.x) {
        int q   = e >> 6;          // K-chunk (4 joints)
        int rr  = (e >> 5) & 1;    // vgpr index
        int l   = e & 31;          // lane
        int col = l & 15;
        int kk  = q * 4 + ((l >= 16) ? 2 : 0) + rr;
        staged[e] = (col < 12) ? Afin[kk][col] : 0.0f;
    }
}

// Kernel 3: streaming LBS. Per wave-iteration: 2 WMMA tiles = 32 points.
// All global addressing uses uniform scalar bases + one 32-bit VGPR offset so the
// backend emits GVS-form (saddr) loads instead of per-lane 64-bit address chains.
__global__ void __launch_bounds__(256) lbs_wmma_kernel(
    const float* __restrict__ points,
    const float* __restrict__ weights,
    const float* __restrict__ ws,
    float* __restrict__ out,
    int N)
{
    const float* staged = ws + 144;
    const int lane = threadIdx.x & 31;
    const int wib  = threadIdx.x >> 5;
    const int wavesPerBlock = blockDim.x >> 5;
    const int waveGlobal = blockIdx.x * wavesPerBlock + wib;
    const int wavesTotal = gridDim.x * wavesPerBlock;

    __shared__ float Tlds[8 * 32 * 17];     // per-wave 32 points x 17 (pad) floats
    float* Tl = Tlds + wib * (32 * 17);

    // Preload all 6 B-operand chunks (held in registers across the whole stream).
    v2f B[6];
#pragma unroll
    for (int q = 0; q < 6; ++q) {
        B[q].x = staged[q * 64 + lane];
        B[q].y = staged[q * 64 + 32 + lane];
    }

    const unsigned uN   = (unsigned)N;
    const int      col  = lane & 15;
    const int      hsel = (lane >> 4) & 1;       // which K pair this half-wave supplies
    const int      prow = hsel ? 8 : 0;
    // divergent (per-lane) part of the weight offset; 24*N*4B < 2^31 so u32 is safe
    const unsigned laneOff = (unsigned)col + (hsel ? 2u * uN : 0u);

    const unsigned Nfull  = uN & ~31u;
    const unsigned stride = (unsigned)wavesTotal * 32u;

    for (unsigned base = (unsigned)waveGlobal * 32u; base < Nfull; base += stride) {
        const unsigned voff = base + laneOff;    // one 32-bit VALU add per iteration
        v8f c0 = {0.f,0.f,0.f,0.f,0.f,0.f,0.f,0.f};
        v8f c1 = {0.f,0.f,0.f,0.f,0.f,0.f,0.f,0.f};
#pragma unroll
        for (int q = 0; q < 6; ++q) {
            const float* wq = weights + (size_t)(4 * q) * uN;   // uniform scalar base per chunk
            v2f a0, a1;
            a0.x = __builtin_nontemporal_load(wq + voff);             // K = 4q + 2*hsel
            a0.y = __builtin_nontemporal_load(wq + voff + uN);        // K = 4q + 2*hsel + 1
            a1.x = __builtin_nontemporal_load(wq + voff + 16u);
            a1.y = __builtin_nontemporal_load(wq + voff + uN + 16u);
            c0 = __builtin_amdgcn_wmma_f32_16x16x4_f32(false, a0, false, B[q], (short)0, c0, false, false);
            c1 = __builtin_amdgcn_wmma_f32_16x16x4_f32(false, a1, false, B[q], (short)0, c1, false, false);
        }
        // D layout: lane (N=col), VGPR r -> M = prow + r. Spill both tiles to LDS.
#pragma unroll
        for (int r = 0; r < 8; ++r) {
            Tl[(prow + r) * 17 + col]      = c0[r];
            Tl[(16 + prow + r) * 17 + col] = c1[r];
        }
        asm volatile("s_wait_dscnt 0" ::: "memory");   // cross-lane LDS RAW within the wave
        // One point per lane: gather its 12 blended coefficients, apply affine.
        const unsigned n = base + (unsigned)lane;
        float px = __builtin_nontemporal_load(points + n);
        float py = __builtin_nontemporal_load(points + n + uN);
        float pz = __builtin_nontemporal_load(points + n + 2u * uN);
        const float* T = Tl + lane * 17;
        float v0 = T[0]*px + T[1]*py + T[2]*pz  + T[3];
        float v1 = T[4]*px + T[5]*py + T[6]*pz  + T[7];
        float v2 = T[8]*px + T[9]*py + T[10]*pz + T[11];
        __builtin_nontemporal_store(v0, out + n);
        __builtin_nontemporal_store(v1, out + n + uN);
        __builtin_nontemporal_store(v2, out + n + 2u * uN);
    }

    // Scalar tail for N not divisible by 32 (N = 2M is divisible; defensive only).
    if (waveGlobal == 0) {
        unsigned n = Nfull + (unsigned)lane;
        if (n < uN) {
            const float* A = ws + 528;
            float m[12];
#pragma unroll
            for (int i = 0; i < 12; ++i) m[i] = 0.f;
            for (int j = 0; j < NJ; ++j) {
                float w = weights[(size_t)j * uN + n];
#pragma unroll
                for (int i = 0; i < 12; ++i) m[i] += w * A[j * 12 + i];
            }
            float px = points[n], py = points[n + uN], pz = points[n + 2u * uN];
            out[n]           = m[0]*px + m[1]*py + m[2]*pz  + m[3];
            out[n + uN]      = m[4]*px + m[5]*py + m[6]*pz  + m[7];
            out[n + 2u * uN] = m[8]*px + m[9]*py + m[10]*pz + m[11];
        }
    }
}

extern "C" void kernel_launch(void* const* d_in, const int* in_sizes, int n_in,
                              void* d_out, int out_size, void* d_ws, size_t ws_size,
                              hipStream_t stream)
{
    const float* points     = (const float*)d_in[0];
    const float* weights    = (const float*)d_in[1];
    const float* beta       = (const float*)d_in[2];
    const float* theta      = (const float*)d_in[3];
    const float* da_theta   = (const float*)d_in[4];
    const float* shapedirs  = (const float*)d_in[5];
    const float* v_template = (const float*)d_in[6];
    const float* Jreg       = (const float*)d_in[7];
    float* out = (float*)d_out;
    float* ws  = (float*)d_ws;
    const int N = in_sizes[0] / 3;

    smpl_joints_kernel<<<24, 256, 0, stream>>>(shapedirs, v_template, Jreg, beta, ws);
    smpl_transforms_kernel<<<1, 64, 0, stream>>>(theta, da_theta, ws);
    lbs_wmma_kernel<<<2048, 256, 0, stream>>>(points, weights, ws, out, N);
}